// FusedSDPAWithKVCache_32040456028273
// MI455X (gfx1250) — compile-verified
//
#include <hip/hip_runtime.h>

typedef __attribute__((ext_vector_type(2))) float v2f;
typedef __attribute__((ext_vector_type(8))) float v8f;

#define NHEADS   16
#define HDIM     128
#define HIDDEN_C 2048
#define BATCH    16
#define QLEN     8
#define TPAST    4088
#define TTOT     4096
#define QK_SCALE 0.08838834764831845f  // 1/sqrt(128)

static __device__ __forceinline__ v8f wmma_f32(v2f a, v2f b, v8f c) {
  // V_WMMA_F32_16X16X4_F32: D(16x16,f32) = A(16x4,f32) x B(4x16,f32) + C
  return __builtin_amdgcn_wmma_f32_16x16x4_f32(
      /*neg_a=*/false, a, /*neg_b=*/false, b,
      /*c_mod=*/(short)0, c, /*reuse_a=*/false, /*reuse_b=*/false);
}

// --- CDNA5 async memory->LDS copy (ASYNCcnt path) --------------------------
// LDS dest VGPR holds workgroup-relative LDS byte offset (= low 32 bits of
// the generic pointer per the LDS aperture mapping); 16 bytes per lane.
static __device__ __forceinline__ void async_load_b128(unsigned lds_byte_off,
                                                       const void* gaddr) {
  asm volatile("global_load_async_to_lds_b128 %0, %1, off"
               :: "v"(lds_byte_off), "v"((unsigned long long)gaddr)
               : "memory");
}
static __device__ __forceinline__ void wait_async0() {
  asm volatile("s_wait_asynccnt 0" ::: "memory");
}
static __device__ __forceinline__ unsigned lds_off(const void* p) {
  return (unsigned)(unsigned long long)p;
}

// ---------------------------------------------------------------------------
// Kernel 1: QKV projection.  X[128 x 2048] (rows = b*8+s) times W^T (W is
// [out,in] row-major).  grid = (32 col-groups, 8 row-tiles, 3 matrices),
// block = 128 (4 waves).  Each wave -> one 16x16 output tile, K = 2048.
// ---------------------------------------------------------------------------
__global__ void qkv_proj_kernel(const float* __restrict__ hs,
                                const float* __restrict__ Wq, const float* __restrict__ bq,
                                const float* __restrict__ Wk, const float* __restrict__ bk,
                                const float* __restrict__ Wv, const float* __restrict__ bv,
                                float* __restrict__ qbuf, float* __restrict__ knew,
                                float* __restrict__ vnew) {
  extern __shared__ float Xs[];  // 16 x 2048 floats = 128 KB
  const int cg   = blockIdx.x;   // 0..31  -> 64 output cols each
  const int rt   = blockIdx.y;   // 0..7   -> 16 rows each
  const int mi   = blockIdx.z;   // 0:Q 1:K 2:V
  const int tid  = threadIdx.x;
  const int lane = tid & 31;
  const int wave = tid >> 5;

  // Stage the 16x2048 X tile into LDS with async B128 copies (no VGPR bounce).
  {
    const char* src   = (const char*)(hs + (size_t)rt * 16 * HIDDEN_C);
    const char* lbase = (const char*)Xs;
    for (int i = tid * 16; i < 16 * HIDDEN_C * 4; i += 128 * 16)
      async_load_b128(lds_off(lbase + i), src + i);
    wait_async0();
  }
  __syncthreads();

  const float* W    = (mi == 0) ? Wq : (mi == 1) ? Wk : Wv;
  const float* bias = (mi == 0) ? bq : (mi == 1) ? bk : bv;

  const int o0    = cg * 64 + wave * 16;  // first output col of this wave
  const int m     = lane & 15;            // A row (== B col index value)
  const int khalf = (lane >> 4) * 2;      // K sub-offset per lane half

  const float* xrow = Xs + m * HIDDEN_C;
  const float* wrow = W + (size_t)(o0 + m) * HIDDEN_C;  // n == lane&15 == m

  v8f acc = {};
  for (int kk = 0; kk < HIDDEN_C / 4; ++kk) {
    const int kb = kk * 4 + khalf;
    v2f a;  a.x = xrow[kb];  a.y = xrow[kb + 1];
    float2 wv = *reinterpret_cast<const float2*>(wrow + kb);
    v2f bf; bf.x = wv.x;  bf.y = wv.y;
    acc = wmma_f32(a, bf, acc);
  }

  // Scatter: D[mm][n], mm = j + 8*(lane>=16), n = lane&15.
  const int n   = lane & 15;
  const int o   = o0 + n;
  const int mhi = (lane >> 4) * 8;
  const int h   = o >> 7;
  const int d   = o & 127;
  const float bv_ = bias[o];
#pragma unroll
  for (int j = 0; j < 8; ++j) {
    const int mm = j + mhi;
    const int r  = rt * 16 + mm;       // global row = b*8 + s
    const int bb = r >> 3, ss = r & 7;
    const float val = acc[j] + bv_;
    const size_t idx = (((size_t)bb * NHEADS + h) * QLEN + ss) * HDIM + d;
    if (mi == 0)      qbuf[idx] = val * QK_SCALE;
    else if (mi == 1) knew[idx] = val;
    else              vnew[idx] = val;
  }
}

// ---------------------------------------------------------------------------
// Kernel 2: fused attention for one (b,h) per workgroup.  8 waves.
//  LDS: Q tile (8x128) + score/prob matrix P (8x4096) + 8 inv-sums.
//  Phase A: scores = Q K^T (+mask) via WMMA, streaming key_cache [D,T].
//  Phase B: row softmax (one wave per row).
//  Phase C: out = P V via WMMA, streaming value_cache [D,T].
// ---------------------------------------------------------------------------
__global__ void attention_kernel(const float* __restrict__ qbuf,
                                 const float* __restrict__ knew,
                                 const float* __restrict__ vnew,
                                 const float* __restrict__ kcache,
                                 const float* __restrict__ vcache,
                                 const float* __restrict__ mask,
                                 float* __restrict__ obuf) {
  extern __shared__ float smem[];
  float* Qs = smem;                     // 8*128
  float* P  = smem + QLEN * HDIM;       // 8*4096
  float* rs = P + QLEN * TTOT;          // 8 inverse row sums

  const int bh   = blockIdx.x;          // b*16 + h
  const int bb   = bh >> 4;
  const int h    = bh & 15;
  const int tid  = threadIdx.x;
  const int lane = tid & 31;
  const int wave = tid >> 5;

  const size_t kvbase  = (size_t)bh * HDIM * (size_t)TPAST;  // cache: row d stride TPAST
  const size_t newbase = (size_t)bh * QLEN * HDIM;           // [t_new, d] row-major

  // Stage Q (8x128 = 4 KB) into LDS: one async B128 per thread.
  async_load_b128(lds_off((const char*)Qs + tid * 16),
                  (const char*)(qbuf + newbase) + tid * 16);
  wait_async0();
  __syncthreads();

  const int m     = lane & 15;
  const int khalf = (lane >> 4) * 2;

  // ---- Phase A: scores -> P ----
  for (int tile = wave; tile < TTOT / 16; tile += 8) {
    const int t0 = tile * 16;
    const int t  = t0 + m;  // this lane's key column (n == m value)

    // prefetch next tile of K for this wave (global_prefetch_b8)
    if (tile + 8 < TTOT / 16) {
      const int tp = t0 + 128 + (lane & 15);
      if (tp < TPAST)
        __builtin_prefetch(kcache + kvbase + (size_t)(lane >> 4) * TPAST + tp, 0, 1);
    }

    v8f acc = {};
    for (int kk = 0; kk < HDIM / 4; ++kk) {
      const int d = kk * 4 + khalf;
      v2f a;
      a.x = (m < QLEN) ? Qs[m * HDIM + d] : 0.0f;
      a.y = (m < QLEN) ? Qs[m * HDIM + d + 1] : 0.0f;
      v2f bf;
      if (t < TPAST) {  // K[d][t], K[d+1][t]  (stride TPAST between d rows)
        const float* kp = kcache + kvbase + (size_t)d * TPAST + t;
        bf.x = kp[0];
        bf.y = kp[TPAST];
      } else {          // new tokens: [t,d] row-major, d even -> float2
        const float* kp = knew + newbase + (size_t)(t - TPAST) * HDIM + d;
        float2 kv2 = *reinterpret_cast<const float2*>(kp);
        bf.x = kv2.x; bf.y = kv2.y;
      }
      acc = wmma_f32(a, bf, acc);
    }
    if (lane < 16) {  // rows 0..7 live in lanes 0..15 (mm = j)
      const float mval = mask[(size_t)bb * TTOT + t0 + lane];
#pragma unroll
      for (int j = 0; j < QLEN; ++j)
        P[j * TTOT + t0 + lane] = acc[j] + mval;
    }
  }
  __syncthreads();

  // ---- Phase B: softmax, wave w owns row w ----
  if (wave < QLEN) {
    float* row = P + wave * TTOT;
    float mx = -3.4e38f;
    for (int i = lane; i < TTOT; i += 32) mx = fmaxf(mx, row[i]);
    for (int off = 16; off > 0; off >>= 1) mx = fmaxf(mx, __shfl_xor(mx, off, 32));
    float sum = 0.0f;
    for (int i = lane; i < TTOT; i += 32) {
      const float e = __expf(row[i] - mx);
      row[i] = e;
      sum += e;
    }
    for (int off = 16; off > 0; off >>= 1) sum += __shfl_xor(sum, off, 32);
    if (lane == 0) rs[wave] = 1.0f / sum;
  }
  __syncthreads();

  // ---- Phase C: out = P * V ; wave w owns d-columns [w*16, w*16+16) ----
  {
    const int d0 = wave * 16;
    const int d  = d0 + m;  // n == m value
    v8f acc = {};
    for (int kk = 0; kk < TTOT / 4; ++kk) {
      const int tb = kk * 4 + khalf;
      v2f a;
      a.x = (m < QLEN) ? P[m * TTOT + tb] : 0.0f;
      a.y = (m < QLEN) ? P[m * TTOT + tb + 1] : 0.0f;
      v2f bf;
      if (tb < TPAST) {  // V[d][tb], V[d][tb+1] contiguous, tb even -> float2
        const float* vp = vcache + kvbase + (size_t)d * TPAST + tb;
        float2 vv = *reinterpret_cast<const float2*>(vp);
        bf.x = vv.x; bf.y = vv.y;
      } else {           // new tokens: [t,d] row-major, stride HDIM between t
        const float* vp = vnew + newbase + (size_t)(tb - TPAST) * HDIM + d;
        bf.x = vp[0];
        bf.y = vp[HDIM];
      }
      acc = wmma_f32(a, bf, acc);
    }
    if (lane < 16) {
#pragma unroll
      for (int j = 0; j < QLEN; ++j)
        obuf[((size_t)bb * QLEN + j) * HIDDEN_C + h * HDIM + d0 + lane] = acc[j] * rs[j];
    }
  }
}

// ---------------------------------------------------------------------------
// Kernel 3: output projection; store as [B, C, 1, S].
// ---------------------------------------------------------------------------
__global__ void out_proj_kernel(const float* __restrict__ X,
                                const float* __restrict__ Wo, const float* __restrict__ bo,
                                float* __restrict__ out) {
  extern __shared__ float Xs[];
  const int cg   = blockIdx.x;
  const int rt   = blockIdx.y;
  const int tid  = threadIdx.x;
  const int lane = tid & 31;
  const int wave = tid >> 5;

  {
    const char* src   = (const char*)(X + (size_t)rt * 16 * HIDDEN_C);
    const char* lbase = (const char*)Xs;
    for (int i = tid * 16; i < 16 * HIDDEN_C * 4; i += 128 * 16)
      async_load_b128(lds_off(lbase + i), src + i);
    wait_async0();
  }
  __syncthreads();

  const int o0    = cg * 64 + wave * 16;
  const int m     = lane & 15;
  const int khalf = (lane >> 4) * 2;

  const float* xrow = Xs + m * HIDDEN_C;
  const float* wrow = Wo + (size_t)(o0 + m) * HIDDEN_C;

  v8f acc = {};
  for (int kk = 0; kk < HIDDEN_C / 4; ++kk) {
    const int kb = kk * 4 + khalf;
    v2f a;  a.x = xrow[kb];  a.y = xrow[kb + 1];
    float2 wv = *reinterpret_cast<const float2*>(wrow + kb);
    v2f bf; bf.x = wv.x;  bf.y = wv.y;
    acc = wmma_f32(a, bf, acc);
  }

  const int n   = lane & 15;
  const int o   = o0 + n;
  const int mhi = (lane >> 4) * 8;
  const float bv = bo[o];
#pragma unroll
  for (int j = 0; j < 8; ++j) {
    const int mm = j + mhi;
    const int r  = rt * 16 + mm;
    const int bb = r >> 3, ss = r & 7;
    // reference output layout [B, C, 1, S]
    out[((size_t)bb * HIDDEN_C + o) * QLEN + ss] = acc[j] + bv;
  }
}

// ---------------------------------------------------------------------------
extern "C" void kernel_launch(void* const* d_in, const int* in_sizes, int n_in,
                              void* d_out, int out_size, void* d_ws, size_t ws_size,
                              hipStream_t stream) {
  const float* hs   = (const float*)d_in[0];
  const float* mask = (const float*)d_in[1];
  const float* kc   = (const float*)d_in[2];
  const float* vc   = (const float*)d_in[3];
  const float* Wq   = (const float*)d_in[4];
  const float* bq   = (const float*)d_in[5];
  const float* Wk   = (const float*)d_in[6];
  const float* bk   = (const float*)d_in[7];
  const float* Wv   = (const float*)d_in[8];
  const float* bv   = (const float*)d_in[9];
  const float* Wo   = (const float*)d_in[10];
  const float* bo   = (const float*)d_in[11];
  float* out = (float*)d_out;

  float* ws   = (float*)d_ws;
  const size_t N = (size_t)BATCH * NHEADS * QLEN * HDIM;  // 262144
  float* qbuf = ws;
  float* knew = ws + N;
  float* vnew = ws + 2 * N;
  float* obuf = ws + 3 * N;  // [B, S, HIDDEN]

  const size_t lds_gemm = (size_t)16 * HIDDEN_C * sizeof(float);                   // 128 KB
  const size_t lds_attn = (size_t)(QLEN * HDIM + QLEN * TTOT + 8) * sizeof(float); // ~132 KB

  qkv_proj_kernel<<<dim3(32, 8, 3), 128, lds_gemm, stream>>>(
      hs, Wq, bq, Wk, bk, Wv, bv, qbuf, knew, vnew);
  attention_kernel<<<dim3(BATCH * NHEADS), 256, lds_attn, stream>>>(
      qbuf, knew, vnew, kc, vc, mask, obuf);
  out_proj_kernel<<<dim3(32, 8), 128, lds_gemm, stream>>>(obuf, Wo, bo, out);
}